// PropagationNetwork_6219112644709
// MI455X (gfx1250) — compile-verified
//
#include <hip/hip_runtime.h>
#include <math.h>

#define CK     64
#define M_TOT  6480   // T*H*W = 4*30*54
#define N_TOT  1620   // H*W   = 30*54
#define KC_TOT 1536   // K*CV  = 3*512
#define TOPK   20

typedef __attribute__((ext_vector_type(2))) float v2f;
typedef __attribute__((ext_vector_type(8))) float v8f;

// ---------------------------------------------------------------------------
// Kernel 1: squared norms of mk columns (per m) and qk columns (per n)
// ---------------------------------------------------------------------------
__global__ void sumsq_kernel(const float* __restrict__ mk, const float* __restrict__ qk,
                             float* __restrict__ msq, float* __restrict__ qsq) {
    int i = blockIdx.x * blockDim.x + threadIdx.x;
    if (i < M_TOT) {
        float s = 0.f;
        #pragma unroll 8
        for (int c = 0; c < CK; ++c) { float v = mk[c * M_TOT + i]; s += v * v; }
        msq[i] = s;
    } else if (i < M_TOT + N_TOT) {
        int n = i - M_TOT;
        float s = 0.f;
        #pragma unroll 8
        for (int c = 0; c < CK; ++c) { float v = qk[c * N_TOT + n]; s += v * v; }
        qsq[n] = s;
    }
}

// ---------------------------------------------------------------------------
// Kernel 2: transpose mv [KC][M] -> mvT [M][KC]  (tiled, LDS, coalesced both sides)
// ---------------------------------------------------------------------------
__global__ void transpose_kernel(const float* __restrict__ mv, float* __restrict__ mvT) {
    __shared__ float t[32][33];
    int tx = threadIdx.x, ty = threadIdx.y;
    int xm = blockIdx.x * 32 + tx;                 // m index (contiguous in src)
    #pragma unroll
    for (int r = 0; r < 32; r += 8) {
        int ykc = blockIdx.y * 32 + ty + r;        // kc index (KC divisible by 32)
        t[ty + r][tx] = (xm < M_TOT) ? mv[(long)ykc * M_TOT + xm] : 0.f;
    }
    __syncthreads();
    int xkc = blockIdx.y * 32 + tx;                // kc (contiguous in dst)
    #pragma unroll
    for (int r = 0; r < 32; r += 8) {
        int ym = blockIdx.x * 32 + ty + r;
        if (ym < M_TOT) mvT[(long)ym * KC_TOT + xkc] = t[tx][ty + r];
    }
}

// ---------------------------------------------------------------------------
// Kernel 3: WMMA f32 GEMM computing transposed affinity:
//   affT[n][m] = (2 * (qk_n . mk_m) - msq[m] - qsq[n]) / 8
// A = qk tile (16 n-rows x 4 k), B = mk tile (4 k x 16 m-cols).
// Each wave: one 16(n) x 32(m) tile => 2 accumulators sharing A.
// ---------------------------------------------------------------------------
__global__ void affinity_wmma_kernel(const float* __restrict__ qk, const float* __restrict__ mk,
                                     const float* __restrict__ msq, const float* __restrict__ qsq,
                                     float* __restrict__ affT) {
    const int NT = (N_TOT + 15) / 16;   // 102 n-tiles (last partial)
    const int MG = (M_TOT + 31) / 32;   // 203 m double-tiles (last partial)
    int waveId = blockIdx.x * (blockDim.x >> 5) + (threadIdx.x >> 5);
    if (waveId >= NT * MG) return;
    int nt = waveId % NT;
    int mg = waveId / NT;
    int lane  = threadIdx.x & 31;
    int lm    = lane & 15;
    int khalf = lane >> 4;              // 0 or 1
    int kh    = khalf << 1;             // k offset 0 or 2
    int n0 = nt * 16;
    int m0 = mg * 32;
    int nrow = min(n0 + lm, N_TOT - 1); // clamped load index for A
    int ma   = min(m0 + lm,      M_TOT - 1);
    int mb   = min(m0 + 16 + lm, M_TOT - 1);

    v8f acc0 = {}; v8f acc1 = {};
    for (int k = 0; k < CK; k += 4) {
        v2f a, b0, b1;
        // A (16x4 f32): lane 0-15 row n, VGPR0 holds k=kh, VGPR1 holds k=kh+1
        a.x  = qk[(k + kh + 0) * N_TOT + nrow];
        a.y  = qk[(k + kh + 1) * N_TOT + nrow];
        // B (4x16 f32): lanes stripe m, VGPR0 k=kh, VGPR1 k=kh+1
        b0.x = mk[(k + kh + 0) * M_TOT + ma];
        b0.y = mk[(k + kh + 1) * M_TOT + ma];
        b1.x = mk[(k + kh + 0) * M_TOT + mb];
        b1.y = mk[(k + kh + 1) * M_TOT + mb];
        acc0 = __builtin_amdgcn_wmma_f32_16x16x4_f32(false, a, false, b0, (short)0, acc0, false, false);
        acc1 = __builtin_amdgcn_wmma_f32_16x16x4_f32(false, a, false, b1, (short)0, acc1, false, false);
    }

    float sma = msq[ma];
    float smb = msq[mb];
    #pragma unroll
    for (int r = 0; r < 8; ++r) {
        int n = n0 + r + (khalf << 3);  // C/D layout: VGPR r -> row r (lanes 0-15) / r+8 (lanes 16-31)
        if (n < N_TOT) {
            float qn = qsq[n];
            if (m0 + lm < M_TOT)
                affT[(long)n * M_TOT + m0 + lm]      = (2.f * acc0[r] - sma - qn) * 0.125f;
            if (m0 + 16 + lm < M_TOT)
                affT[(long)n * M_TOT + m0 + 16 + lm] = (2.f * acc1[r] - smb - qn) * 0.125f;
        }
    }
}

// ---------------------------------------------------------------------------
// Kernel 4: per-column top-20 + softmax. One wave per column (wave32).
// Lane-private sorted top-20 in registers, LDS dump, 32-sorted-heads merge
// via wave shuffles, then softmax of the 20 (sorted-desc) values.
// ---------------------------------------------------------------------------
__global__ void topk_softmax_kernel(const float* __restrict__ affT,
                                    float* __restrict__ wgt, int* __restrict__ tidx) {
    __shared__ float lv[8 * 32 * TOPK];
    __shared__ int   li[8 * 32 * TOPK];
    int wave = threadIdx.x >> 5;
    int lane = threadIdx.x & 31;
    int n = blockIdx.x * 8 + wave;
    if (n >= N_TOT) return;                       // whole-wave exit, no block barriers used

    float val[TOPK]; int idx[TOPK];
    #pragma unroll
    for (int j = 0; j < TOPK; ++j) { val[j] = -INFINITY; idx[j] = 0; }

    // coalesced scan: lanes stride consecutive m in affT[n][*]
    for (int m = lane; m < M_TOT; m += 32) {
        float v = affT[(long)n * M_TOT + m];
        if (v > val[TOPK - 1]) {
            val[TOPK - 1] = v; idx[TOPK - 1] = m;
            #pragma unroll
            for (int j = TOPK - 1; j > 0; --j) {   // single bubble pass restores sorted order
                if (val[j] > val[j - 1]) {
                    float tv = val[j]; val[j] = val[j - 1]; val[j - 1] = tv;
                    int   ti = idx[j]; idx[j] = idx[j - 1]; idx[j - 1] = ti;
                }
            }
        }
    }

    int base = (wave * 32 + lane) * TOPK;
    #pragma unroll
    for (int j = 0; j < TOPK; ++j) { lv[base + j] = val[j]; li[base + j] = idx[j]; }

    // merge 32 sorted lists: 20 rounds of wave-argmax over list heads
    int hp = 0;
    float res_v = 0.f; int res_i = 0;
    for (int r = 0; r < TOPK; ++r) {
        float hv = (hp < TOPK) ? lv[base + hp] : -INFINITY;
        int   hi = (hp < TOPK) ? li[base + hp] : 0;
        float bv = hv; int bl = lane;
        #pragma unroll
        for (int off = 16; off > 0; off >>= 1) {
            float ov = __shfl_xor(bv, off, 32);
            int   ol = __shfl_xor(bl, off, 32);
            if (ov > bv) { bv = ov; bl = ol; }
        }
        int wi = __shfl(hi, bl, 32);
        if (lane == bl) hp++;
        if (lane == r) { res_v = bv; res_i = wi; }
    }

    // softmax over the 20 (res_v on lane r is the r-th largest; lane 0 has the max)
    float v0 = __shfl(res_v, 0, 32);
    float e = (lane < TOPK) ? __expf(res_v - v0) : 0.f;
    float s = e;
    #pragma unroll
    for (int off = 16; off > 0; off >>= 1) s += __shfl_xor(s, off, 32);
    if (lane < TOPK) {
        wgt[n * TOPK + lane]  = e / s;
        tidx[n * TOPK + lane] = res_i;
    }
}

// ---------------------------------------------------------------------------
// Kernel 5: sparse readout. Tile = 64 kc x 32 n per block (256 threads).
// Reads mvT rows (contiguous over kc -> coalesced, L2-resident); output
// transposed through LDS so stores are coalesced over n.
// ---------------------------------------------------------------------------
__global__ void readout_kernel(const float* __restrict__ mvT, const float* __restrict__ wgt,
                               const int* __restrict__ tidx, float* __restrict__ out) {
    __shared__ float sw[32 * TOPK];
    __shared__ int   si[32 * TOPK];
    __shared__ float so[64][33];

    int n0  = blockIdx.y * 32;
    int kc0 = blockIdx.x * 64;
    int tid = threadIdx.x;

    for (int i = tid; i < 32 * TOPK; i += 256) {
        int nl = i / TOPK;
        int n  = n0 + nl;
        sw[i] = (n < N_TOT) ? wgt[n * TOPK + (i % TOPK)] : 0.f;
        si[i] = (n < N_TOT) ? tidx[n * TOPK + (i % TOPK)] : 0;
    }
    __syncthreads();

    int kc  = kc0 + (tid & 63);       // lanes cover consecutive kc -> coalesced mvT reads
    int ng4 = tid >> 6;               // 4 groups x 8 n each
    float acc[8];
    #pragma unroll
    for (int q = 0; q < 8; ++q) acc[q] = 0.f;

    #pragma unroll
    for (int q = 0; q < 8; ++q) {
        int nl = ng4 * 8 + q;
        #pragma unroll
        for (int j = 0; j < TOPK; ++j) {
            acc[q] += sw[nl * TOPK + j] * mvT[(long)si[nl * TOPK + j] * KC_TOT + kc];
        }
    }

    #pragma unroll
    for (int q = 0; q < 8; ++q) so[tid & 63][ng4 * 8 + q] = acc[q];
    __syncthreads();

    for (int i = tid; i < 64 * 32; i += 256) {
        int kcl = i >> 5;
        int nl  = i & 31;
        int n   = n0 + nl;
        if (n < N_TOT) out[(long)(kc0 + kcl) * N_TOT + n] = so[kcl][nl];
    }
}

// ---------------------------------------------------------------------------
extern "C" void kernel_launch(void* const* d_in, const int* in_sizes, int n_in,
                              void* d_out, int out_size, void* d_ws, size_t ws_size,
                              hipStream_t stream) {
    const float* mk = (const float*)d_in[0];   // [64][6480]
    const float* qk = (const float*)d_in[1];   // [64][1620]
    const float* mv = (const float*)d_in[2];   // [1536][6480]
    float* out = (float*)d_out;                // [1536][1620]

    // workspace layout (~82.1 MB total)
    float* aff = (float*)d_ws;                           // affT [N][M]
    float* mvT = aff + (size_t)M_TOT * N_TOT;            // [M][KC]
    float* msq = mvT + (size_t)M_TOT * KC_TOT;           // [M]
    float* qsq = msq + M_TOT;                            // [N]
    float* wgt = qsq + N_TOT;                            // [N][20]
    int*  tIdx = (int*)(wgt + (size_t)N_TOT * TOPK);     // [N][20]

    sumsq_kernel<<<(M_TOT + N_TOT + 255) / 256, 256, 0, stream>>>(mk, qk, msq, qsq);

    transpose_kernel<<<dim3((M_TOT + 31) / 32, KC_TOT / 32), dim3(32, 8), 0, stream>>>(mv, mvT);

    const int NT = (N_TOT + 15) / 16;   // 102
    const int MG = (M_TOT + 31) / 32;   // 203
    int totalWaves = NT * MG;
    affinity_wmma_kernel<<<(totalWaves + 7) / 8, 256, 0, stream>>>(qk, mk, msq, qsq, aff);

    topk_softmax_kernel<<<(N_TOT + 7) / 8, 256, 0, stream>>>(aff, wgt, tIdx);

    readout_kernel<<<dim3(KC_TOT / 64, (N_TOT + 31) / 32), 256, 0, stream>>>(mvT, wgt, tIdx, out);
}